// Net_65927747993607
// MI455X (gfx1250) — compile-verified
//
#include <hip/hip_runtime.h>
#include <utility>

// ---------------------------------------------------------------------------
// 1-2-3-GNN forward for MI455X (gfx1250).
// Memory-bound edge scatter in fp32 HW atomics (index loaded once per edge,
// fully-unrolled float4 gather clause); dense node GEMMs via
// V_WMMA_F32_16X16X4_F32 (exact fp32 matrix pipe on CDNA5).
// ---------------------------------------------------------------------------

typedef __attribute__((ext_vector_type(2))) float v2f;
typedef __attribute__((ext_vector_type(8))) float v8f;

#if defined(__HIP_DEVICE_COMPILE__)
#define FATOMIC_ADD(p, v) unsafeAtomicAdd((p), (v))
#else
#define FATOMIC_ADD(p, v) atomicAdd((p), (v))
#endif

// -------------------------------- utility kernels --------------------------

__global__ void zero_f32(float* __restrict__ p, long n) {
  long i = (long)blockIdx.x * blockDim.x + threadIdx.x;
  long stride = (long)gridDim.x * blockDim.x;
  for (; i < n; i += stride) p[i] = 0.0f;
}

// agg[dst[e]] += x[src[e]]  -- one thread per edge, K templated so the
// K/4 float4 gathers issue as one clause and indices load exactly once.
template <int K>
__global__ void scatter_add_edges(const float* __restrict__ x,
                                  const int* __restrict__ src,
                                  const int* __restrict__ dst,
                                  float* __restrict__ agg, int E) {
  int e = blockIdx.x * blockDim.x + threadIdx.x;
  if (e >= E) return;
  int s = src[e], d = dst[e];
  const float4* xr = (const float4*)(x + (long)s * K);
  float* o = agg + (long)d * K;
  float4 v[K / 4];
#pragma unroll
  for (int c = 0; c < K / 4; ++c) v[c] = xr[c];
#pragma unroll
  for (int c = 0; c < K / 4; ++c) {
    FATOMIC_ADD(o + 4 * c + 0, v[c].x);
    FATOMIC_ADD(o + 4 * c + 1, v[c].y);
    FATOMIC_ADD(o + 4 * c + 2, v[c].z);
    FATOMIC_ADD(o + 4 * c + 3, v[c].w);
  }
}

// out[idx[n]] += x[n]   (K = 64 fixed)
__global__ void scatter_add_rows64(const float* __restrict__ x,
                                   const int* __restrict__ idx,
                                   float* __restrict__ out, int N) {
  int n = blockIdx.x * blockDim.x + threadIdx.x;
  if (n >= N) return;
  int d = idx[n];
  const float4* xr = (const float4*)(x + (long)n * 64);
  float* o = out + (long)d * 64;
  float4 v[16];
#pragma unroll
  for (int c = 0; c < 16; ++c) v[c] = xr[c];
#pragma unroll
  for (int c = 0; c < 16; ++c) {
    FATOMIC_ADD(o + 4 * c + 0, v[c].x);
    FATOMIC_ADD(o + 4 * c + 1, v[c].y);
    FATOMIC_ADD(o + 4 * c + 2, v[c].z);
    FATOMIC_ADD(o + 4 * c + 3, v[c].w);
  }
}

__global__ void scatter_count(const int* __restrict__ idx,
                              float* __restrict__ cnt, int N) {
  int n = blockIdx.x * blockDim.x + threadIdx.x;
  if (n < N) FATOMIC_ADD(&cnt[idx[n]], 1.0f);
}

// xcat[b][colOff + k] = sum[b][k] / max(cnt[b], 1)
__global__ void mean_to_xcat(const float* __restrict__ sum,
                             const float* __restrict__ cnt,
                             float* __restrict__ xcat, int B, int K,
                             int colOff, int ld) {
  int t = blockIdx.x * blockDim.x + threadIdx.x;
  if (t >= B * K) return;
  int b = t / K, k = t - b * K;
  float c = cnt[b];
  c = c > 1.0f ? c : 1.0f;
  xcat[(long)b * ld + colOff + k] = sum[(long)b * K + k] / c;
}

// out[n] = [ sum[n]/max(cnt,1) (K wide) | iso[n] (KI wide) ]
__global__ void pool_concat(const float* __restrict__ sum,
                            const float* __restrict__ cnt,
                            const float* __restrict__ iso,
                            float* __restrict__ out, int N, int K, int KI) {
  int W = K + KI;
  long t = (long)blockIdx.x * blockDim.x + threadIdx.x;
  if (t >= (long)N * W) return;
  int n = (int)(t / W), k = (int)(t - (long)n * W);
  float v;
  if (k < K) {
    float c = cnt[n];
    c = c > 1.0f ? c : 1.0f;
    v = sum[(long)n * K + k] / c;
  } else {
    v = iso[(long)n * KI + (k - K)];
  }
  out[(long)n * W + k] = v;
}

// --------------------------- WMMA fused linear ------------------------------
// out[N, NC] = A1[N,K1] @ W1[K1,NC] (+ A2[N,K2] @ W2[K2,NC]) + bias, opt. ELU.
// One wave per 16x16 tile; block handles one 16-row strip (NC/16 waves).
// N % 16 == 0, K1/K2 % 4 == 0, NC in {32, 64}. Uses V_WMMA_F32_16X16X4_F32.
__global__ __launch_bounds__(128) void fused_linear_wmma(
    const float* __restrict__ A1, const float* __restrict__ W1, int K1,
    const float* __restrict__ A2, const float* __restrict__ W2, int K2,
    const float* __restrict__ bias, float* __restrict__ out, int NC,
    int applyElu) {
  __shared__ float lds[4384];  // fits 16*(193) + 16*(81)
  const int tid = threadIdx.x;
  const int lane = tid & 31;
  const int wave = tid >> 5;
  const int row0 = blockIdx.x << 4;
  const int nthreads = blockDim.x;

  const int s1 = K1 + 1;  // padded stride: kills LDS bank conflicts
  for (int i = tid; i < 16 * K1; i += nthreads) {
    int r = i / K1, c = i - r * K1;
    lds[r * s1 + c] = A1[(long)(row0 + r) * K1 + c];
  }
  float* lds2 = lds + 16 * s1;
  const int s2 = K2 + 1;
  if (K2 > 0) {
    for (int i = tid; i < 16 * K2; i += nthreads) {
      int r = i / K2, c = i - r * K2;
      lds2[r * s2 + c] = A2[(long)(row0 + r) * K2 + c];
    }
  }
  __syncthreads();

  // 16x16x4 f32 WMMA lane layout:
  //   A (16x4): lane m = row; half (lane>=16) selects K pair {0,1} vs {2,3}
  //   B (4x16): lane n = col; same K pairing
  //   C/D (16x16): acc[i] -> row i + 8*half, col = lane&15
  const int m = lane & 15;
  const int half = lane >> 4;
  const int koff = half * 2;
  const int col0 = wave << 4;

  v8f acc = {0.f, 0.f, 0.f, 0.f, 0.f, 0.f, 0.f, 0.f};
  for (int kk = 0; kk < K1; kk += 4) {
    v2f a, b;
    a.x = lds[m * s1 + kk + koff];
    a.y = lds[m * s1 + kk + koff + 1];
    b.x = W1[(long)(kk + koff) * NC + col0 + m];
    b.y = W1[(long)(kk + koff + 1) * NC + col0 + m];
    acc = __builtin_amdgcn_wmma_f32_16x16x4_f32(false, a, false, b, (short)0,
                                                acc, false, false);
  }
  for (int kk = 0; kk < K2; kk += 4) {
    v2f a, b;
    a.x = lds2[m * s2 + kk + koff];
    a.y = lds2[m * s2 + kk + koff + 1];
    b.x = W2[(long)(kk + koff) * NC + col0 + m];
    b.y = W2[(long)(kk + koff + 1) * NC + col0 + m];
    acc = __builtin_amdgcn_wmma_f32_16x16x4_f32(false, a, false, b, (short)0,
                                                acc, false, false);
  }

  float bv = bias ? bias[col0 + m] : 0.0f;
#pragma unroll
  for (int i = 0; i < 8; ++i) {
    int r = row0 + i + half * 8;
    float v = acc[i] + bv;
    if (applyElu) v = v > 0.0f ? v : (__expf(v) - 1.0f);
    out[(long)r * NC + col0 + m] = v;
  }
}

// final K->C linear + log_softmax, one thread per graph; fully unrolled
template <int K, int C>
__global__ void fc3_logsoftmax(const float* __restrict__ h,
                               const float* __restrict__ W,
                               const float* __restrict__ b,
                               float* __restrict__ out, int B) {
  int row = blockIdx.x * blockDim.x + threadIdx.x;
  if (row >= B) return;
  float hr[K];
#pragma unroll
  for (int k = 0; k < K; ++k) hr[k] = h[(long)row * K + k];
  float logits[C];
  float mx = -1e30f;
#pragma unroll
  for (int c = 0; c < C; ++c) {
    float s = b[c];
#pragma unroll
    for (int k = 0; k < K; ++k) s += hr[k] * W[(long)k * C + c];
    logits[c] = s;
    mx = s > mx ? s : mx;
  }
  float sum = 0.0f;
#pragma unroll
  for (int c = 0; c < C; ++c) sum += __expf(logits[c] - mx);
  float lse = mx + __logf(sum);
#pragma unroll
  for (int c = 0; c < C; ++c) out[(long)row * C + c] = logits[c] - lse;
}

// ------------------------------- host driver --------------------------------

struct ConvP {
  const float* Wl;
  const float* Wr;
  const float* br;
};

static void launch_scatter_edges(const float* x, const int* src, const int* dst,
                                 float* agg, int E, int K, hipStream_t stream) {
  int blocks = (E + 255) / 256;
  switch (K) {
    case 32:
      scatter_add_edges<32><<<blocks, 256, 0, stream>>>(x, src, dst, agg, E);
      break;
    case 80:
      scatter_add_edges<80><<<blocks, 256, 0, stream>>>(x, src, dst, agg, E);
      break;
    default:
      scatter_add_edges<64><<<blocks, 256, 0, stream>>>(x, src, dst, agg, E);
      break;
  }
}

extern "C" void kernel_launch(void* const* d_in, const int* in_sizes, int n_in,
                              void* d_out, int out_size, void* d_ws,
                              size_t ws_size, hipStream_t stream) {
  (void)n_in; (void)out_size; (void)ws_size;
  const int N1 = in_sizes[0] / 32;
  const int N2 = in_sizes[1] / 16;
  const int N3 = in_sizes[2] / 16;
  const int E1 = in_sizes[39] / 2;
  const int E2 = in_sizes[40] / 2;
  const int E3 = in_sizes[41] / 2;
  const int A2E = in_sizes[42] / 2;  // 2*N2
  const int A3E = in_sizes[43] / 2;  // 3*N3

  const float* x = (const float*)d_in[0];
  const float* iso2 = (const float*)d_in[1];
  const float* iso3 = (const float*)d_in[2];
  // params in JAX pytree order (sorted dict keys): conv dicts -> {Wl, Wr, br}
  const float* P[36];
  for (int i = 0; i < 36; ++i) P[i] = (const float*)d_in[3 + i];
  ConvP ci = {P[0], P[1], P[2]};
  ConvP cv[3][3];
  {
    int idx = 3;
    for (int j = 0; j < 3; ++j)
      for (int i = 0; i < 3; ++i) {
        cv[j][i] = {P[idx], P[idx + 1], P[idx + 2]};
        idx += 3;
      }
  }
  const float* fcW[3] = {P[30], P[32], P[34]};
  const float* fcB[3] = {P[31], P[33], P[35]};
  const int* ei[3] = {(const int*)d_in[39], (const int*)d_in[40],
                      (const int*)d_in[41]};
  const int* as2 = (const int*)d_in[42];
  const int* as3 = (const int*)d_in[43];
  const int* bat[3] = {(const int*)d_in[44], (const int*)d_in[45],
                       (const int*)d_in[46]};

  // workspace layout (floats)
  float* ws = (float*)d_ws;
  const long region = (long)N3 * 80;
  float* R0 = ws;
  float* AGG = ws + region;
  float* R2 = ws + 2 * region;
  float* small = ws + 3 * region;
  float* cnt = small;              // up to N3
  float* bsum = cnt + N3;          // 256*64
  float* bcnt = bsum + 256 * 64;   // 256
  float* xcat = bcnt + 256;        // 256*192
  float* fc1 = xcat + 256 * 192;   // 256*64
  float* fc2 = fc1 + 256 * 64;     // 256*32

  auto zero = [&](float* p, long n) {
    int blocks = (int)((n + 255) / 256);
    if (blocks > 2048) blocks = 2048;
    zero_f32<<<blocks, 256, 0, stream>>>(p, n);
  };
  auto conv = [&](const float* hin, int Kin, const int* e, int E,
                  const ConvP& c, float* hout, int N) {
    zero(AGG, (long)N * Kin);
    launch_scatter_edges(hin, e, e + E, AGG, E, Kin, stream);
    fused_linear_wmma<<<N / 16, 128, 0, stream>>>(AGG, c.Wr, Kin, hin, c.Wl,
                                                  Kin, c.br, hout, 64, 1);
  };
  auto batch_pool = [&](const float* h, const int* b, int N, int colOff) {
    zero(bsum, 256 * 64);
    zero(bcnt, 256);
    scatter_add_rows64<<<(N + 255) / 256, 256, 0, stream>>>(h, b, bsum, N);
    scatter_count<<<(N + 255) / 256, 256, 0, stream>>>(b, bcnt, N);
    mean_to_xcat<<<(256 * 64 + 255) / 256, 256, 0, stream>>>(bsum, bcnt, xcat,
                                                             256, 64, colOff,
                                                             192);
  };
  auto assign_concat = [&](const float* h, const int* as, int AE,
                           const float* iso, int Nout, float* outBuf) {
    zero(AGG, (long)Nout * 64);
    zero(cnt, Nout);
    launch_scatter_edges(h, as, as + AE, AGG, AE, 64, stream);
    scatter_count<<<(AE + 255) / 256, 256, 0, stream>>>(as + AE, cnt, AE);
    long tot = (long)Nout * 80;
    pool_concat<<<(int)((tot + 255) / 256), 256, 0, stream>>>(AGG, cnt, iso,
                                                              outBuf, Nout, 64,
                                                              16);
  };

  float* hin = R0;
  float* hout = R2;

  // conv_initial: h = elu(agg(x)@Wr + br + x@Wl)
  zero(AGG, (long)N1 * 32);
  launch_scatter_edges(x, ei[0], ei[0] + E1, AGG, E1, 32, stream);
  fused_linear_wmma<<<N1 / 16, 128, 0, stream>>>(AGG, ci.Wr, 32, x, ci.Wl, 32,
                                                 ci.br, hin, 64, 1);
  // level 1
  for (int i = 0; i < 3; ++i) {
    conv(hin, 64, ei[0], E1, cv[0][i], hout, N1);
    std::swap(hin, hout);
  }
  batch_pool(hin, bat[0], N1, 0);
  assign_concat(hin, as2, A2E, iso2, N2, hout);
  std::swap(hin, hout);
  // level 2
  conv(hin, 80, ei[1], E2, cv[1][0], hout, N2); std::swap(hin, hout);
  conv(hin, 64, ei[1], E2, cv[1][1], hout, N2); std::swap(hin, hout);
  conv(hin, 64, ei[1], E2, cv[1][2], hout, N2); std::swap(hin, hout);
  batch_pool(hin, bat[1], N2, 64);
  assign_concat(hin, as3, A3E, iso3, N3, hout);
  std::swap(hin, hout);
  // level 3
  conv(hin, 80, ei[2], E3, cv[2][0], hout, N3); std::swap(hin, hout);
  conv(hin, 64, ei[2], E3, cv[2][1], hout, N3); std::swap(hin, hout);
  conv(hin, 64, ei[2], E3, cv[2][2], hout, N3); std::swap(hin, hout);
  batch_pool(hin, bat[2], N3, 128);

  // head: [256,192] -> 64 -> 32 -> 10 + log_softmax
  fused_linear_wmma<<<256 / 16, 128, 0, stream>>>(xcat, fcW[0], 192, nullptr,
                                                  nullptr, 0, fcB[0], fc1, 64,
                                                  1);
  fused_linear_wmma<<<256 / 16, 64, 0, stream>>>(fc1, fcW[1], 64, nullptr,
                                                 nullptr, 0, fcB[1], fc2, 32,
                                                 1);
  fc3_logsoftmax<32, 10><<<1, 256, 0, stream>>>(fc2, fcW[2], fcB[2],
                                                (float*)d_out, 256);
}